// VectorQuantizer_2241972928757
// MI455X (gfx1250) — compile-verified
//
#include <hip/hip_runtime.h>
#include <hip/hip_bf16.h>

typedef __attribute__((ext_vector_type(2))) float v2f;
typedef __attribute__((ext_vector_type(8))) float v8f;

#define NUM_EMB   1024
#define EMB_DIM   128
#define N_VEC     32768      // 2*16*32*32
#define SPATIAL   16384      // 16*32*32 (per batch)
#define BATCH_STR 2097152    // 128*16384 (elements per batch in NCDHW)
#define OUT_ELEMS 4194304    // 2*128*16*32*32

// ---------------------------------------------------------------------------
// Kernel 1a: build pair-interleaved, pre-scaled transposed codebook:
//   etp[(c/2)*NUM_EMB + k] = float2(-2*e[k][2p], -2*e[k][2p+1])
// Also zero the loss accumulator. Writes are fully coalesced (k fastest).
// ---------------------------------------------------------------------------
__global__ void vq_prep_transpose(const float* __restrict__ emb,
                                  float2* __restrict__ etp,
                                  float* __restrict__ lossAcc) {
    int t = blockIdx.x * blockDim.x + threadIdx.x;   // 65536 threads
    if (t == 0) *lossAcc = 0.0f;
    if (t < NUM_EMB * (EMB_DIM / 2)) {
        int p = t >> 10;                 // channel pair 0..63
        int k = t & (NUM_EMB - 1);       // embedding index (fastest -> coalesced write)
        float a = emb[k * EMB_DIM + 2 * p];
        float b = emb[k * EMB_DIM + 2 * p + 1];
        etp[p * NUM_EMB + k] = make_float2(-2.0f * a, -2.0f * b);
    }
}

// ---------------------------------------------------------------------------
// Kernel 1b: enorm[k] = ||e_k||^2   (one wave per codebook row)
// ---------------------------------------------------------------------------
__global__ void vq_prep_enorm(const float* __restrict__ emb,
                              float* __restrict__ enorm) {
    int k = blockIdx.x;
    int lane = threadIdx.x;                 // blockDim = 32 (one wave32)
    const float* row = emb + k * EMB_DIM;
    float s = 0.0f;
    #pragma unroll
    for (int j = 0; j < 4; ++j) {
        float v = row[lane + 32 * j];
        s += v * v;
    }
    #pragma unroll
    for (int m = 16; m >= 1; m >>= 1) s += __shfl_xor(s, m, 32);
    if (lane == 0) enorm[k] = s;
}

// ---------------------------------------------------------------------------
// Kernel 2: WMMA f32 argmin. One wave handles 16 rows of x against K=1024.
//   A (16x4 f32) layout: lanes 0-15 / 16-31 = M 0-15; vgpr0 = K{0|2}, vgpr1 = K{1|3}
//   C/D (16x16 f32): lanes 0-15 N=lane (M=r); lanes 16-31 N=lane-16 (M=r+8)
// etp is pre-scaled by -2, so score = ||e_k||^2 + acc  (||x||^2 const per row).
// One global_load_b64 per v_wmma in the hot loop.
// ---------------------------------------------------------------------------
__global__ void __launch_bounds__(256)
vq_argmin_kernel(const float* __restrict__ x,       // NCDHW input
                 const v2f*  __restrict__ etp,      // [C/2][K] float2, scaled by -2
                 const float* __restrict__ enorm,   // [K]
                 int* __restrict__ idx_out) {
    const int wave = (blockIdx.x * blockDim.x + threadIdx.x) >> 5; // 0..2047
    const int lane = threadIdx.x & 31;
    const int half = lane >> 4;       // 0: lanes 0-15, 1: lanes 16-31
    const int lm   = lane & 15;

    const int n0      = wave << 4;                 // first of 16 rows
    const int b       = n0 >> 14;                  // tile never crosses batch
    const int spatial = n0 - (b << 14);
    const int base    = b * BATCH_STR + spatial;   // + c*SPATIAL + lm

    // Load all A fragments once: 32 fragments covering C = 0..127 (4 per frag)
    v2f a[32];
    #pragma unroll
    for (int f = 0; f < 32; ++f) {
        const int c0 = 4 * f + 2 * half;
        a[f].x = x[base + (c0    ) * SPATIAL + lm];
        a[f].y = x[base + (c0 + 1) * SPATIAL + lm];
    }

    float minv[8];
    int   mini[8];
    #pragma unroll
    for (int r = 0; r < 8; ++r) { minv[r] = 3.4e38f; mini[r] = 0; }

    for (int kt = 0; kt < NUM_EMB / 16; ++kt) {    // 64 column tiles
        const int k = kt * 16 + lm;                // embedding index for this lane
        v8f acc = {};
        #pragma unroll
        for (int f = 0; f < 32; ++f) {
            // B fragment: rows (c0, c0+1) for this half-wave = pair index 2f+half
            const v2f bf = etp[(2 * f + half) * NUM_EMB + k];   // single b64 load
            acc = __builtin_amdgcn_wmma_f32_16x16x4_f32(
                /*neg_a=*/false, a[f], /*neg_b=*/false, bf,
                /*c_mod=*/(short)0, acc, /*reuse_a=*/false, /*reuse_b=*/false);
        }
        const float en = enorm[k];
        #pragma unroll
        for (int r = 0; r < 8; ++r) {
            float v = en + acc[r];                 // ||e||^2 - 2 x.e (scale folded)
            if (v < minv[r]) { minv[r] = v; mini[r] = k; }  // strict < keeps first k
        }
    }

    // Reduce across the 16 lanes of each half-wave (same row, different k).
    #pragma unroll
    for (int r = 0; r < 8; ++r) {
        float v = minv[r];
        int   i = mini[r];
        #pragma unroll
        for (int m = 8; m >= 1; m >>= 1) {
            float ov = __shfl_xor(v, m, 32);
            int   oi = __shfl_xor(i, m, 32);
            if (ov < v || (ov == v && oi < i)) { v = ov; i = oi; }  // first-index ties
        }
        if (lm == 0) idx_out[n0 + r + 8 * half] = i;
    }
}

// ---------------------------------------------------------------------------
// Kernel 3: gather quantized output (NCDHW) + accumulate squared-diff loss.
// ---------------------------------------------------------------------------
__global__ void __launch_bounds__(256)
vq_output_kernel(const float* __restrict__ x,
                 const float* __restrict__ emb,
                 const int* __restrict__ idx,
                 float* __restrict__ out,
                 float* __restrict__ lossAcc) {
    const int o = blockIdx.x * 256 + threadIdx.x;      // flat NCDHW index
    const int spatial = o & (SPATIAL - 1);
    const int c = (o >> 14) & 127;
    const int b = o >> 21;
    const int n = (b << 14) + spatial;

    const float q = emb[idx[n] * EMB_DIM + c];
    const float d = q - x[o];
    out[o] = q;                                        // straight-through == q

    // block reduction of d*d
    float s = d * d;
    #pragma unroll
    for (int m = 16; m >= 1; m >>= 1) s += __shfl_xor(s, m, 32);
    __shared__ float sm[8];
    const int lane = threadIdx.x & 31;
    const int w    = threadIdx.x >> 5;
    if (lane == 0) sm[w] = s;
    __syncthreads();
    if (w == 0) {
        s = (lane < 8) ? sm[lane] : 0.0f;
        #pragma unroll
        for (int m = 4; m >= 1; m >>= 1) s += __shfl_xor(s, m, 32);
        if (lane == 0) atomicAdd(lossAcc, s);
    }
}

// ---------------------------------------------------------------------------
// Kernel 4: finalize loss = (1 + COMMITMENT_COST) * mean(diff^2)
// ---------------------------------------------------------------------------
__global__ void vq_finalize(const float* __restrict__ lossAcc,
                            float* __restrict__ out_loss) {
    if (threadIdx.x == 0 && blockIdx.x == 0)
        *out_loss = 1.25f * (*lossAcc) * (1.0f / (float)OUT_ELEMS);
}

// ---------------------------------------------------------------------------
extern "C" void kernel_launch(void* const* d_in, const int* in_sizes, int n_in,
                              void* d_out, int out_size, void* d_ws, size_t ws_size,
                              hipStream_t stream) {
    const float* x   = (const float*)d_in[0];   // [2,128,16,32,32] fp32
    const float* emb = (const float*)d_in[1];   // [1024,128] fp32
    float* out = (float*)d_out;                 // 4194304 quantized; loss stored at out[2097152]

    // workspace layout (floats)
    float*  ws      = (float*)d_ws;
    float2* etp     = (float2*)ws;         // 131072 f : pair-interleaved -2*e^T
    float*  enorm   = ws + 131072;         //   1024 f : ||e_k||^2
    float*  lossAcc = ws + 132096;         //      1 f : loss accumulator
    int*    idx     = (int*)(ws + 132128); //  32768 i : argmin indices

    vq_prep_transpose<<<(NUM_EMB * (EMB_DIM / 2) + 255) / 256, 256, 0, stream>>>(emb, etp, lossAcc);
    vq_prep_enorm   <<<NUM_EMB, 32, 0, stream>>>(emb, enorm);
    vq_argmin_kernel<<<N_VEC / 16 / 8, 256, 0, stream>>>(x, (const v2f*)etp, enorm, idx); // 256 blocks, 8 waves
    vq_output_kernel<<<OUT_ELEMS / 256, 256, 0, stream>>>(x, emb, idx, out, lossAcc);
    vq_finalize     <<<1, 32, 0, stream>>>(lossAcc, out + 2097152);
}